// SelfAttention_39015482917537
// MI455X (gfx1250) — compile-verified
//
#include <hip/hip_runtime.h>
#include <hip/hip_bf16.h>

typedef __attribute__((ext_vector_type(2))) float v2f;
typedef __attribute__((ext_vector_type(8))) float v8f;

#define BB 2
#define CC 64
#define LL 8192
#define MC 2048          // K/V LDS staging chunk (floats)
#define THREADS 256
#define WAVES 8          // 256 / wave32
#define ROWS_PER_BLOCK 128  // 8 waves * 16 rows

#define LOG2E 1.4426950408889634f

// ---------------------------------------------------------------------------
// Phase 1: pointwise 1x1 conv: q/k/v[b,l] = dot(W, x[b,:,l]) + bias
// ---------------------------------------------------------------------------
__global__ void qkv_kernel(const float* __restrict__ x,
                           const float* __restrict__ Wq, const float* __restrict__ Wqb,
                           const float* __restrict__ Wk, const float* __restrict__ Wkb,
                           const float* __restrict__ Wv, const float* __restrict__ Wvb,
                           float* __restrict__ Qb, float* __restrict__ Kb,
                           float* __restrict__ Vb) {
    int b = blockIdx.y;
    int l = blockIdx.x * THREADS + threadIdx.x;
    const float* xb = x + (size_t)b * CC * LL + l;
    float qa = 0.f, ka = 0.f, va = 0.f;
#pragma unroll
    for (int c = 0; c < CC; ++c) {
        float xv = xb[(size_t)c * LL];          // coalesced across lanes
        qa = fmaf(Wq[c], xv, qa);               // W[c] uniform -> s_load
        ka = fmaf(Wk[c], xv, ka);
        va = fmaf(Wv[c], xv, va);
    }
    Qb[b * LL + l] = qa + Wqb[0];
    Kb[b * LL + l] = ka + Wkb[0];
    Vb[b * LL + l] = va + Wvb[0];
}

// ---------------------------------------------------------------------------
// Phase 2: per-batch K min / max (for the rank-1 softmax max trick)
// ---------------------------------------------------------------------------
__global__ void kstat_kernel(const float* __restrict__ Kb, float* __restrict__ kstat) {
    __shared__ float smax[THREADS], smin[THREADS];
    int b = blockIdx.x;
    int tid = threadIdx.x;
    float mx = -3.4e38f, mn = 3.4e38f;
    for (int i = tid; i < LL; i += THREADS) {
        float k = Kb[b * LL + i];
        mx = fmaxf(mx, k);
        mn = fminf(mn, k);
    }
    smax[tid] = mx; smin[tid] = mn;
    __syncthreads();
    for (int s = THREADS / 2; s > 0; s >>= 1) {
        if (tid < s) {
            smax[tid] = fmaxf(smax[tid], smax[tid + s]);
            smin[tid] = fminf(smin[tid], smin[tid + s]);
        }
        __syncthreads();
    }
    if (tid == 0) { kstat[b * 2 + 0] = smax[0]; kstat[b * 2 + 1] = smin[0]; }
}

// ---------------------------------------------------------------------------
// Phase 3: rank-1 softmax attention via WMMA f32 16x16x4.
// Wave w owns rows l0..l0+15. Per m-chunk of 4:
//   A (16x4)  = exp2(qs_l * k_m - cs_l)         -> 2 VGPRs/lane (v2f)
//   B (4x16)  = col0: 1.0, col1: v_m, else 0    -> 2 VGPRs/lane (v2f)
//   C += A*B  : col0 accumulates denom, col1 accumulates numerator.
// qs = q*log2e, cs = rowmax*log2e are prescaled once, so each element costs
// exactly one v_fma + one v_exp_f32; the WMMA reduction co-executes on the
// matrix pipe with the TRANS exp stream.
// ---------------------------------------------------------------------------
__global__ void attn_kernel(const float* __restrict__ Qb, const float* __restrict__ Kb,
                            const float* __restrict__ Vb, const float* __restrict__ kstat,
                            float* __restrict__ Obuf) {
    __shared__ float sK[MC];
    __shared__ float sV[MC];
    __shared__ float sred[WAVES][16][2];

    int b    = blockIdx.x / (LL / ROWS_PER_BLOCK);
    int tile = blockIdx.x % (LL / ROWS_PER_BLOCK);
    int tid  = threadIdx.x;
    int wave = tid >> 5;          // wave32
    int lane = tid & 31;
    int row  = lane & 15;         // M index (A) / N index (B)
    int half = lane >> 4;         // K-half: lanes 16-31 hold K=2,3
    int koff = half * 2;

    int l0 = tile * ROWS_PER_BLOCK + wave * 16;
    float q = Qb[b * LL + l0 + row];
    float kmax = kstat[b * 2 + 0];
    float kmin = kstat[b * 2 + 1];
    float crow = (q >= 0.f) ? q * kmax : q * kmin;   // exact row max of scores

    // Prescale by log2(e): exp(q*k - c) == exp2(qs*k - cs)
    float qs = q * LOG2E;
    float ncs = -(crow * LOG2E);

    // B-fragment build constants: col0 -> 1.0, col1 -> v, rest -> 0
    float base  = (row == 0) ? 1.0f : 0.0f;
    float vmask = (row == 1) ? 1.0f : 0.0f;

    const float* Kg = Kb + b * LL;
    const float* Vg = Vb + b * LL;

    v8f acc = {};
    for (int m0 = 0; m0 < LL; m0 += MC) {
        __syncthreads();   // previous chunk fully consumed
        for (int i = tid * 4; i < MC; i += THREADS * 4) {
            *(float4*)(sK + i) = *(const float4*)(Kg + m0 + i);
            *(float4*)(sV + i) = *(const float4*)(Vg + m0 + i);
        }
        __syncthreads();
#pragma unroll 8
        for (int m = 0; m < MC; m += 4) {
            float k0 = sK[m + koff];
            float k1 = sK[m + koff + 1];
            float v0 = sV[m + koff];
            float v1 = sV[m + koff + 1];
            float a0 = __builtin_amdgcn_exp2f(fmaf(qs, k0, ncs));
            float a1 = __builtin_amdgcn_exp2f(fmaf(qs, k1, ncs));
            v2f A = {a0, a1};
            v2f Bf = {fmaf(vmask, v0, base), fmaf(vmask, v1, base)};
            acc = __builtin_amdgcn_wmma_f32_16x16x4_f32(
                false, A, false, Bf, (short)0, acc, false, false);
        }
    }

    // D layout: VGPR r, lanes 0-15 -> M=r, lanes 16-31 -> M=r+8, N = lane&15.
    // Columns 0 (denom) and 1 (numer) -> LDS, then 16 lanes finalize.
    if (row < 2) {
#pragma unroll
        for (int r = 0; r < 8; ++r)
            sred[wave][r + half * 8][row] = acc[r];
    }
    __syncthreads();
    if (lane < 16) {
        float denom = sred[wave][lane][0];
        float numer = sred[wave][lane][1];
        Obuf[b * LL + l0 + lane] = numer / denom;
    }
}

// ---------------------------------------------------------------------------
// Phase 4: out[b,c,l] = beta * (Wo_w[c] * O[b,l] + Wo_b[c]) + x[b,c,l]
// ---------------------------------------------------------------------------
__global__ void out_kernel(const float* __restrict__ x,
                           const float* __restrict__ Wo_w, const float* __restrict__ Wo_b,
                           const float* __restrict__ beta, const float* __restrict__ Obuf,
                           float* __restrict__ out) {
    int idx = blockIdx.x * THREADS + threadIdx.x;   // B*C*L total
    int l  = idx % LL;
    int bc = idx / LL;
    int c  = bc % CC;
    int b  = bc / CC;
    float o = Obuf[b * LL + l];
    out[idx] = beta[0] * fmaf(Wo_w[c], o, Wo_b[c]) + x[idx];
}

// ---------------------------------------------------------------------------
extern "C" void kernel_launch(void* const* d_in, const int* in_sizes, int n_in,
                              void* d_out, int out_size, void* d_ws, size_t ws_size,
                              hipStream_t stream) {
    const float* x    = (const float*)d_in[0];
    const float* Wq_w = (const float*)d_in[1];
    const float* Wq_b = (const float*)d_in[2];
    const float* Wk_w = (const float*)d_in[3];
    const float* Wk_b = (const float*)d_in[4];
    const float* Wv_w = (const float*)d_in[5];
    const float* Wv_b = (const float*)d_in[6];
    const float* Wo_w = (const float*)d_in[7];
    const float* Wo_b = (const float*)d_in[8];
    const float* beta = (const float*)d_in[9];
    float* out = (float*)d_out;

    float* ws    = (float*)d_ws;
    float* Qb    = ws;                    // B*L
    float* Kb    = ws + 1 * BB * LL;      // B*L
    float* Vb    = ws + 2 * BB * LL;      // B*L
    float* Obuf  = ws + 3 * BB * LL;      // B*L
    float* kstat = ws + 4 * BB * LL;      // 2*B

    dim3 g1(LL / THREADS, BB);
    qkv_kernel<<<g1, THREADS, 0, stream>>>(x, Wq_w, Wq_b, Wk_w, Wk_b, Wv_w, Wv_b,
                                           Qb, Kb, Vb);
    kstat_kernel<<<BB, THREADS, 0, stream>>>(Kb, kstat);
    attn_kernel<<<BB * (LL / ROWS_PER_BLOCK), THREADS, 0, stream>>>(Qb, Kb, Vb, kstat, Obuf);
    out_kernel<<<(BB * CC * LL) / THREADS, THREADS, 0, stream>>>(x, Wo_w, Wo_b, beta,
                                                                 Obuf, out);
}